// GraphSAGE_86268713107998
// MI455X (gfx1250) — compile-verified
//
#include <hip/hip_runtime.h>
#include <hip/hip_bf16.h>

typedef __attribute__((ext_vector_type(2))) float v2f;
typedef __attribute__((ext_vector_type(8))) float v8f;

#define N_NODES 50000
#define N_EDGES 1600000
#define DFEAT   128
#define LDS_STRIDE 132   // 128 + 4 pad -> conflict-free ds_load_b64 across 16 lanes

// Non-returning f32 atomic add (L2 atomic unit, STOREcnt-tracked).
__device__ __forceinline__ void atomic_add_f32(float* addr, float val) {
    asm volatile("global_atomic_add_f32 %0, %1, off" : : "v"(addr), "v"(val) : "memory");
}

__global__ __launch_bounds__(256) void zero_kernel(float* __restrict__ p, int n) {
    int i = blockIdx.x * 256 + threadIdx.x;
    if (i < n) p[i] = 0.0f;
}

// One wave32 per edge: lane l moves floats [4l, 4l+3] of the 128-wide row.
__global__ __launch_bounds__(256)
void sage_scatter_kernel(const float* __restrict__ x,
                         const int* __restrict__ src,
                         const int* __restrict__ dst,
                         float* __restrict__ agg,
                         float* __restrict__ deg,
                         int nEdges) {
    const int e = (blockIdx.x << 3) + (threadIdx.x >> 5);
    if (e >= nEdges) return;
    const int lane = threadIdx.x & 31;
    const int s = src[e];
    const int d = dst[e];
    const float4 v = *(const float4*)(x + (size_t)s * DFEAT + (lane << 2));
    float* p = agg + (size_t)d * DFEAT + (lane << 2);
    atomic_add_f32(p + 0, v.x);
    atomic_add_f32(p + 1, v.y);
    atomic_add_f32(p + 2, v.z);
    atomic_add_f32(p + 3, v.w);
    if (lane == 0) atomic_add_f32(deg + d, 1.0f);
}

// out[16 rows] = relu( (agg/deg) @ Wl^T + x @ Wr^T + b )
// Block: 16 output rows, 8 waves -> 8 column tiles of 16.
// A/B fragments per ISA 7.12.2 f32 layouts; 64 x v_wmma_f32_16x16x4_f32 per wave.
__global__ __launch_bounds__(256)
void sage_gemm_kernel(const float* __restrict__ xin,
                      const float* __restrict__ agg,
                      const float* __restrict__ deg,
                      const float* __restrict__ Wl,
                      const float* __restrict__ Wr,
                      const float* __restrict__ bias,
                      float* __restrict__ out) {
    __shared__ float lsA[16 * LDS_STRIDE];   // mean tile (pre-scaled by 1/deg)
    __shared__ float lsX[16 * LDS_STRIDE];   // x tile

    const int row0 = blockIdx.x << 4;

    // Stage 16x128 tiles of mean and x into LDS (float4 per thread, 2 iters).
    for (int t = threadIdx.x; t < 512; t += 256) {
        const int r  = t >> 5;          // row 0..15
        const int c4 = (t & 31) << 2;   // col 0..124 step 4
        const float dinv = 1.0f / fmaxf(deg[row0 + r], 1.0f);
        float4 va = *(const float4*)(agg + (size_t)(row0 + r) * DFEAT + c4);
        float4 vx = *(const float4*)(xin + (size_t)(row0 + r) * DFEAT + c4);
        va.x *= dinv; va.y *= dinv; va.z *= dinv; va.w *= dinv;
        *(float4*)(lsA + r * LDS_STRIDE + c4) = va;   // (r*132+c4)*4 % 16 == 0
        *(float4*)(lsX + r * LDS_STRIDE + c4) = vx;
    }
    __syncthreads();   // all global reads of xin done before any store to out (in-place safe)

    const int lane  = threadIdx.x & 31;
    const int wave  = threadIdx.x >> 5;
    const int col0  = wave << 4;
    const int n     = lane & 15;           // A: row M ; B: col N ; C: col N
    const int khalf = (lane >> 4) << 1;    // K sub-pair: lanes 0-15 -> {0,1}, 16-31 -> {2,3}

    const float* la = lsA + n * LDS_STRIDE + khalf;
    const float* lx = lsX + n * LDS_STRIDE + khalf;
    const float* bl = Wl + (size_t)(col0 + n) * DFEAT + khalf;  // B[k][n] = W[n][k]
    const float* br = Wr + (size_t)(col0 + n) * DFEAT + khalf;

    v8f acc = {};
#pragma unroll
    for (int k = 0; k < DFEAT; k += 4) {
        v2f a = *(const v2f*)(la + k);
        v2f b = *(const v2f*)(bl + k);
        acc = __builtin_amdgcn_wmma_f32_16x16x4_f32(false, a, false, b,
                                                    (short)0, acc, false, false);
    }
#pragma unroll
    for (int k = 0; k < DFEAT; k += 4) {
        v2f a = *(const v2f*)(lx + k);
        v2f b = *(const v2f*)(br + k);
        acc = __builtin_amdgcn_wmma_f32_16x16x4_f32(false, a, false, b,
                                                    (short)0, acc, false, false);
    }

    // C layout: VGPR r -> row (r + 8*(lane>=16)), col n. Bias + ReLU + store.
    const float bj = bias[col0 + n];
    const int mbase = (lane >> 4) << 3;
    float* orow = out + (size_t)(row0 + mbase) * DFEAT + col0 + n;
#pragma unroll
    for (int r = 0; r < 8; ++r) {
        float v = acc[r] + bj;
        orow[(size_t)r * DFEAT] = fmaxf(v, 0.0f);
    }
}

extern "C" void kernel_launch(void* const* d_in, const int* in_sizes, int n_in,
                              void* d_out, int out_size, void* d_ws, size_t ws_size,
                              hipStream_t stream) {
    const float* x   = (const float*)d_in[0];
    const int*   ei  = (const int*)d_in[1];
    const float* W1l = (const float*)d_in[2];
    const float* W1r = (const float*)d_in[3];
    const float* b1  = (const float*)d_in[4];
    const float* W2l = (const float*)d_in[5];
    const float* W2r = (const float*)d_in[6];
    const float* b2  = (const float*)d_in[7];
    float* out = (float*)d_out;

    const int* src = ei;             // edge_index[0]
    const int* dst = ei + N_EDGES;   // edge_index[1]

    float* agg = (float*)d_ws;                       // [N, 128]
    float* deg = agg + (size_t)N_NODES * DFEAT;      // [N]

    const int zn = N_NODES * DFEAT + N_NODES;
    dim3 zgrid((zn + 255) / 256);
    dim3 sgrid((N_EDGES + 7) / 8);
    dim3 ggrid(N_NODES / 16);       // 3125, exact

    // ---- layer 1: h1 (stored in d_out) = relu(sage(x)) ----
    zero_kernel<<<zgrid, 256, 0, stream>>>(agg, zn);
    sage_scatter_kernel<<<sgrid, 256, 0, stream>>>(x, src, dst, agg, deg, N_EDGES);
    sage_gemm_kernel<<<ggrid, 256, 0, stream>>>(x, agg, deg, W1l, W1r, b1, out);

    // ---- layer 2: out = relu(sage(h1)), in-place (LDS-staged tiles make it safe) ----
    zero_kernel<<<zgrid, 256, 0, stream>>>(agg, zn);
    sage_scatter_kernel<<<sgrid, 256, 0, stream>>>(out, src, dst, agg, deg, N_EDGES);
    sage_gemm_kernel<<<ggrid, 256, 0, stream>>>(out, agg, deg, W2l, W2r, b2, out);
}